// IPAPooling_17282948399405
// MI455X (gfx1250) — compile-verified
//
#include <hip/hip_runtime.h>
#include <math.h>

#define B_    2
#define N_    1024
#define CS_   384
#define H_    12
#define C_    16
#define PQ_   4
#define PV_   8
#define T_    1538      // NPAD_ + TP_
#define NPAD_ 1026
#define TP_   512
#define BT_   (B_ * T_)
#define OCD_  576       // H*C + H*PV*3 + H*PV = 192+288+96

typedef float v2f __attribute__((ext_vector_type(2)));
typedef float v8f __attribute__((ext_vector_type(8)));

// ---------------------------------------------------------------- WMMA fp32
__device__ __forceinline__ v8f wmma4(v2f a, v2f b, v8f c) {
  // D = A(16x4) * B(4x16) + C(16x16), fp32 tensor op on gfx1250
  return __builtin_amdgcn_wmma_f32_16x16x4_f32(false, a, false, b, (short)0, c,
                                               false, false);
}

__device__ __forceinline__ int imin(int a, int b) { return a < b ? a : b; }

// ---------------------------------------------------------------- prep
__device__ __forceinline__ int padIdx(int pi) {
  int i = pi - 1;
  return i < 0 ? 0 : (i > N_ - 1 ? N_ - 1 : i);
}

__global__ void prep_kernel(const float* __restrict__ trans,
                            const float* __restrict__ rots,
                            const float* __restrict__ s,
                            float* __restrict__ s_all,
                            float* __restrict__ t_all,
                            float* __restrict__ r_all) {
  int bt = blockIdx.x;          // [0, B*T)
  int b = bt / T_;
  int t = bt % T_;
  int n0, n1 = -1, n2 = -1;
  if (t < NPAD_) {
    n0 = padIdx(t);
  } else {
    int p = t - NPAD_;
    n0 = padIdx(2 * p);
    n1 = padIdx(2 * p + 1);
    n2 = padIdx(2 * p + 2);
  }
  bool pool = (n1 >= 0);
  const float third = 1.0f / 3.0f;

  const float* sb = s + (size_t)b * N_ * CS_;
  float* so = s_all + (size_t)bt * CS_;
  for (int c = threadIdx.x; c < CS_; c += blockDim.x) {
    float val = sb[(size_t)n0 * CS_ + c];
    if (pool)
      val = (val + sb[(size_t)n1 * CS_ + c] + sb[(size_t)n2 * CS_ + c]) * third;
    so[c] = val;
  }
  if (threadIdx.x < 9) {
    int c = threadIdx.x;
    const float* rb = rots + (size_t)b * N_ * 9;
    float val = rb[(size_t)n0 * 9 + c];
    if (pool) val = (val + rb[(size_t)n1 * 9 + c] + rb[(size_t)n2 * 9 + c]) * third;
    r_all[(size_t)bt * 9 + c] = val;
  }
  if (threadIdx.x < 3) {
    int c = threadIdx.x;
    const float* tb = trans + (size_t)b * N_ * 3;
    float val = tb[(size_t)n0 * 3 + c];
    if (pool) val = (val + tb[(size_t)n1 * 3 + c] + tb[(size_t)n2 * 3 + c]) * third;
    t_all[(size_t)bt * 3 + c] = val;
  }
}

// ---------------------------------------------------------------- generic WMMA GEMM
// C[M,NN] = A[M,K] @ W[K,NN], row-major.  One wave computes a 32x64 output
// slab: 2 M-tiles x 4 N-tiles, so each B fragment feeds two WMMAs.
// NN is a compile-time constant -> B loads use immediate offsets and
// running pointers (no per-iteration 64-bit multiplies).
// Out-of-range rows are clamped (duplicate data), stores are guarded.
template <int NN>
__global__ void gemm16t(const float* __restrict__ A, const float* __restrict__ W,
                        float* __restrict__ Cc, int M, int K) {
  constexpr int TN4 = (NN + 63) / 64;
  int tm = blockIdx.x / TN4;
  int tn4 = blockIdx.x % TN4;
  int lane = threadIdx.x;
  int lh = lane >> 4;           // lane-half (K-split for A/B fragments)
  int ll = lane & 15;

  int ar0 = imin(tm * 32 + ll, M - 1);        // clamp: loads stay in bounds
  int ar1 = imin(tm * 32 + 16 + ll, M - 1);
  const float* ap0 = A + (size_t)ar0 * K + 2 * lh;
  const float* ap1 = A + (size_t)ar1 * K + 2 * lh;

  const float* wpj[4];                         // per-lane running W pointers
#pragma unroll
  for (int j = 0; j < 4; ++j) {
    int bc = imin(tn4 * 64 + j * 16 + ll, NN - 1);
    wpj[j] = W + (size_t)(2 * lh) * NN + bc;
  }

  v8f acc0[4], acc1[4];
#pragma unroll
  for (int j = 0; j < 4; ++j) {
    acc0[j] = (v8f){0, 0, 0, 0, 0, 0, 0, 0};
    acc1[j] = (v8f){0, 0, 0, 0, 0, 0, 0, 0};
  }

#pragma unroll 2
  for (int k0 = 0; k0 < K; k0 += 4) {
    v2f a0 = *(const v2f*)(ap0 + k0);          // merged b64 loads
    v2f a1 = *(const v2f*)(ap1 + k0);
#pragma unroll
    for (int j = 0; j < 4; ++j) {
      v2f bf;
      bf.x = wpj[j][0];                        // imm-offset pair loads
      bf.y = wpj[j][NN];
      acc0[j] = wmma4(a0, bf, acc0[j]);
      acc1[j] = wmma4(a1, bf, acc1[j]);
      wpj[j] += 4 * NN;
    }
  }
#pragma unroll
  for (int j = 0; j < 4; ++j) {
    int cn = tn4 * 64 + j * 16 + ll;
    if (cn < NN) {
#pragma unroll
      for (int vv = 0; vv < 8; ++vv) {         // C layout: VGPR vv -> M = vv + 8*half
        int r0 = tm * 32 + vv + 8 * lh;
        if (r0 < M) Cc[(size_t)r0 * NN + cn] = acc0[j][vv];
        int r1 = tm * 32 + 16 + vv + 8 * lh;
        if (r1 < M) Cc[(size_t)r1 * NN + cn] = acc1[j][vv];
      }
    }
  }
}

// ---------------------------------------------------------------- rigid transform (in place)
__global__ void to_global_kernel(float* __restrict__ pts, int P,
                                 const float* __restrict__ t_all,
                                 const float* __restrict__ r_all,
                                 float* __restrict__ sqOut) {
  int idx = blockIdx.x * blockDim.x + threadIdx.x;  // (b,t,h)
  if (idx >= BT_ * H_) return;
  int bt = idx / H_;
  const float* R = r_all + (size_t)bt * 9;
  const float* tt = t_all + (size_t)bt * 3;
  float* p = pts + (size_t)idx * P * 3;
  float ssq = 0.0f;
  for (int pi = 0; pi < P; ++pi) {
    float x = p[pi * 3 + 0], y = p[pi * 3 + 1], z = p[pi * 3 + 2];
    float gx = R[0] * x + R[1] * y + R[2] * z + tt[0];
    float gy = R[3] * x + R[4] * y + R[5] * z + tt[1];
    float gz = R[6] * x + R[7] * y + R[8] * z + tt[2];
    p[pi * 3 + 0] = gx; p[pi * 3 + 1] = gy; p[pi * 3 + 2] = gz;
    ssq += gx * gx + gy * gy + gz * gz;
  }
  if (sqOut) sqOut[idx] = ssq;
}

// ---------------------------------------------------------------- attention
__device__ __forceinline__ v8f compute_scores(
    int b, int h, int j0, int ll, int lh, const v2f* aq, const v2f* aqg,
    const float* sqr, const float* __restrict__ kbuf,
    const float* __restrict__ kgbuf, const float* __restrict__ skbuf,
    float gcoef) {
  v8f cqk = {0, 0, 0, 0, 0, 0, 0, 0};
  v8f cpp = {0, 0, 0, 0, 0, 0, 0, 0};
  int j = j0 + ll;
  int jc = imin(j, T_ - 1);                      // clamp: loads in bounds
  size_t rb = ((size_t)(b * T_ + jc) * H_ + h);
  const v2f* krow = (const v2f*)(kbuf + rb * C_);
#pragma unroll
  for (int c0 = 0; c0 < 4; ++c0) {     // sum over C=16 in 4 K-chunks
    v2f bf = krow[c0 * 2 + lh];
    cqk = wmma4(aq[c0], bf, cqk);
  }
  const v2f* kgrow = (const v2f*)(kgbuf + rb * (PQ_ * 3));
#pragma unroll
  for (int d0 = 0; d0 < 3; ++d0) {     // sum over PQ*3=12 dims in 3 chunks
    v2f bf = kgrow[d0 * 2 + lh];
    cpp = wmma4(aqg[d0], bf, cpp);
  }
  float skj = skbuf[rb];
  bool jv = (j < T_);
  v8f sc;
#pragma unroll
  for (int vv = 0; vv < 8; ++vv) {
    float s = 0.70710678f * 0.25f * cqk[vv]                 // w_l * logit/sqrt(C)
              - gcoef * (sqr[vv] + skj - 2.0f * cpp[vv]);   // - g*w_c/2 * d2
    sc[vv] = jv ? s : -1e30f;                               // mask -> P = 0
  }
  return sc;
}

__global__ void attn_kernel(const float* __restrict__ qb,
                            const float* __restrict__ kb,
                            const float* __restrict__ vbuf,
                            const float* __restrict__ qg,
                            const float* __restrict__ kg,
                            const float* __restrict__ vg,
                            const float* __restrict__ sqb,
                            const float* __restrict__ skb,
                            const float* __restrict__ gamma,
                            float* __restrict__ o, float* __restrict__ opg) {
  __shared__ float Plds[16][17];
  int blk = blockIdx.x;                 // b*H*32 + h*32 + qt
  int qt = blk & 31;
  int h = (blk >> 5) % H_;
  int b = blk / (32 * H_);
  int lane = threadIdx.x;
  int lh = lane >> 4;
  int ll = lane & 15;
  int i0 = qt * 16;                     // pooled-query tile base

  float gm = gamma[h];
  float sp = (gm > 20.0f) ? gm : log1pf(expf(gm));  // softplus
  float gcoef = sp * 0.11785113f;                   // * w_c/2, w_c = sqrt(1/18)

  // Hoisted A fragments (key-loop invariant). A layout: lanes=M, K split by half.
  int tq = NPAD_ + i0 + ll;
  size_t qrow = ((size_t)(b * T_ + tq) * H_ + h);
  const v2f* qrp = (const v2f*)(qb + qrow * C_);
  v2f aq[4];
#pragma unroll
  for (int c0 = 0; c0 < 4; ++c0) aq[c0] = qrp[c0 * 2 + lh];
  const v2f* qgp = (const v2f*)(qg + qrow * 12);
  v2f aqg[3];
#pragma unroll
  for (int d0 = 0; d0 < 3; ++d0) aqg[d0] = qgp[d0 * 2 + lh];
  float sqr[8];
#pragma unroll
  for (int vv = 0; vv < 8; ++vv)
    sqr[vv] = sqb[(size_t)(b * T_ + NPAD_ + i0 + vv + 8 * lh) * H_ + h];

  // ---- pass 1: row max
  float rm[8];
#pragma unroll
  for (int vv = 0; vv < 8; ++vv) rm[vv] = -1e30f;
  for (int j0 = 0; j0 < T_; j0 += 16) {
    v8f sc = compute_scores(b, h, j0, ll, lh, aq, aqg, sqr, kb, kg, skb, gcoef);
#pragma unroll
    for (int vv = 0; vv < 8; ++vv) rm[vv] = fmaxf(rm[vv], sc[vv]);
  }
#pragma unroll
  for (int off = 1; off < 16; off <<= 1) {
#pragma unroll
    for (int vv = 0; vv < 8; ++vv)
      rm[vv] = fmaxf(rm[vv], __shfl_xor(rm[vv], off, 32));
  }

  // ---- pass 2: P = exp(s - max), accumulate P@v, P@vg, rowsum
  v8f co = {0, 0, 0, 0, 0, 0, 0, 0};
  v8f cg1 = {0, 0, 0, 0, 0, 0, 0, 0};
  v8f cg2 = {0, 0, 0, 0, 0, 0, 0, 0};
  float ps[8];
#pragma unroll
  for (int vv = 0; vv < 8; ++vv) ps[vv] = 0.0f;

  for (int j0 = 0; j0 < T_; j0 += 16) {
    v8f sc = compute_scores(b, h, j0, ll, lh, aq, aqg, sqr, kb, kg, skb, gcoef);
    v8f P;
#pragma unroll
    for (int vv = 0; vv < 8; ++vv) {
      float e = expf(sc[vv] - rm[vv]);   // masked cols: exp(-1e30 - m) == 0
      P[vv] = e;
      ps[vv] += e;
    }
    __syncthreads();
#pragma unroll
    for (int vv = 0; vv < 8; ++vv) Plds[vv + 8 * lh][ll] = P[vv];  // C layout -> tile
    __syncthreads();
#pragma unroll
    for (int kk = 0; kk < 4; ++kk) {
      v2f aP;                             // re-read P in A layout
      aP.x = Plds[ll][kk * 4 + 2 * lh];
      aP.y = Plds[ll][kk * 4 + 2 * lh + 1];
      int jr = j0 + kk * 4 + 2 * lh;      // B fragment key rows jr, jr+1
      size_t r0 = ((size_t)(b * T_ + imin(jr, T_ - 1)) * H_ + h);
      size_t r1 = ((size_t)(b * T_ + imin(jr + 1, T_ - 1)) * H_ + h);
      v2f bv;
      bv.x = vbuf[r0 * C_ + ll];
      bv.y = vbuf[r1 * C_ + ll];
      co = wmma4(aP, bv, co);
      v2f bg1;
      bg1.x = vg[r0 * 24 + ll];
      bg1.y = vg[r1 * 24 + ll];
      cg1 = wmma4(aP, bg1, cg1);
      int llc = 16 + (ll & 7);            // cols >= 8 duplicate col 0..7; never stored
      v2f bg2;
      bg2.x = vg[r0 * 24 + llc];
      bg2.y = vg[r1 * 24 + llc];
      cg2 = wmma4(aP, bg2, cg2);
    }
  }
#pragma unroll
  for (int off = 1; off < 16; off <<= 1) {
#pragma unroll
    for (int vv = 0; vv < 8; ++vv) ps[vv] += __shfl_xor(ps[vv], off, 32);
  }
  // normalize + store (only pooled queries exist here)
#pragma unroll
  for (int vv = 0; vv < 8; ++vv) {
    float inv = 1.0f / ps[vv];
    int iq = i0 + vv + 8 * lh;
    size_t ob = ((size_t)(b * TP_ + iq) * H_ + h);
    o[ob * C_ + ll] = co[vv] * inv;
    opg[ob * 24 + ll] = cg1[vv] * inv;
    if (ll < 8) opg[ob * 24 + 16 + ll] = cg2[vv] * inv;
  }
}

// ---------------------------------------------------------------- build oc = [o, opl, norms]
__global__ void build_oc_kernel(const float* __restrict__ o,
                                const float* __restrict__ opg,
                                const float* __restrict__ t_all,
                                const float* __restrict__ r_all,
                                float* __restrict__ oc) {
  int idx = blockIdx.x * blockDim.x + threadIdx.x;  // (b,q,h)
  if (idx >= B_ * TP_ * H_) return;
  int h = idx % H_;
  int bq = idx / H_;
  int qq = bq % TP_;
  int b = bq / TP_;
  int bt = b * T_ + NPAD_ + qq;
  const float* R = r_all + (size_t)bt * 9;
  const float* tt = t_all + (size_t)bt * 3;
  float* ocrow = oc + (size_t)bq * OCD_;
  const float* orow = o + (size_t)idx * C_;
  for (int c = 0; c < C_; ++c) ocrow[h * C_ + c] = orow[c];
  const float* prow = opg + (size_t)idx * 24;
  for (int p = 0; p < PV_; ++p) {
    float dx = prow[p * 3 + 0] - tt[0];
    float dy = prow[p * 3 + 1] - tt[1];
    float dz = prow[p * 3 + 2] - tt[2];
    float lx = R[0] * dx + R[3] * dy + R[6] * dz;   // R^T * d
    float ly = R[1] * dx + R[4] * dy + R[7] * dz;
    float lz = R[2] * dx + R[5] * dy + R[8] * dz;
    int pb = 192 + (h * PV_ + p) * 3;
    ocrow[pb + 0] = lx; ocrow[pb + 1] = ly; ocrow[pb + 2] = lz;
    ocrow[480 + h * PV_ + p] = sqrtf(lx * lx + ly * ly + lz * lz + 1e-8f);
  }
}

// ---------------------------------------------------------------- update + quaternion compose
__global__ void final_kernel(const float* __restrict__ su,
                             const float* __restrict__ wupd,
                             const float* __restrict__ bupd,
                             const float* __restrict__ t_all,
                             const float* __restrict__ r_all,
                             float* __restrict__ out) {
  int row = blockIdx.x;          // b*TP + q
  int lane = threadIdx.x;
  float acc[6];
#pragma unroll
  for (int j = 0; j < 6; ++j) acc[j] = 0.0f;
  const float* srow = su + (size_t)row * CS_;
  for (int k = lane; k < CS_; k += 32) {
    float sv = srow[k];
#pragma unroll
    for (int j = 0; j < 6; ++j) acc[j] += sv * wupd[k * 6 + j];
  }
#pragma unroll
  for (int off = 16; off > 0; off >>= 1) {
#pragma unroll
    for (int j = 0; j < 6; ++j) acc[j] += __shfl_xor(acc[j], off, 32);
  }
  if (lane == 0) {
    float u[6];
#pragma unroll
    for (int j = 0; j < 6; ++j) u[j] = acc[j] + bupd[j];
    float qx = u[0], qy = u[1], qz = u[2];
    float inv = 1.0f / sqrtf(1.0f + qx * qx + qy * qy + qz * qz);
    float qw = inv; qx *= inv; qy *= inv; qz *= inv;
    float ru[9] = {1 - 2 * (qy * qy + qz * qz), 2 * (qx * qy - qw * qz), 2 * (qx * qz + qw * qy),
                   2 * (qx * qy + qw * qz), 1 - 2 * (qx * qx + qz * qz), 2 * (qy * qz - qw * qx),
                   2 * (qx * qz - qw * qy), 2 * (qy * qz + qw * qx), 1 - 2 * (qx * qx + qy * qy)};
    int b = row / TP_;
    int qq = row % TP_;
    int bt = b * T_ + NPAD_ + qq;
    const float* R = r_all + (size_t)bt * 9;
    const float* tt = t_all + (size_t)bt * 3;
#pragma unroll
    for (int i = 0; i < 3; ++i)
      out[(size_t)row * 3 + i] =
          R[i * 3 + 0] * u[3] + R[i * 3 + 1] * u[4] + R[i * 3 + 2] * u[5] + tt[i];
    float* ro = out + (size_t)B_ * TP_ * 3 + (size_t)row * 9;
#pragma unroll
    for (int i = 0; i < 3; ++i)
#pragma unroll
      for (int kk = 0; kk < 3; ++kk)
        ro[i * 3 + kk] = R[i * 3 + 0] * ru[0 + kk] + R[i * 3 + 1] * ru[3 + kk] +
                         R[i * 3 + 2] * ru[6 + kk];
  }
}

// ---------------------------------------------------------------- launch helpers
template <int NN>
static inline void launch_gemm(const float* A, const float* W, float* Cc, int M,
                               int K, hipStream_t stream) {
  constexpr int TN4 = (NN + 63) / 64;
  int tiles = ((M + 31) / 32) * TN4;
  gemm16t<NN><<<tiles, 32, 0, stream>>>(A, W, Cc, M, K);
}

// ---------------------------------------------------------------- launch
extern "C" void kernel_launch(void* const* d_in, const int* in_sizes, int n_in,
                              void* d_out, int out_size, void* d_ws, size_t ws_size,
                              hipStream_t stream) {
  const float* trans = (const float*)d_in[0];
  const float* rots  = (const float*)d_in[1];
  const float* s     = (const float*)d_in[2];
  const float* wq    = (const float*)d_in[3];
  const float* wk    = (const float*)d_in[4];
  const float* wv    = (const float*)d_in[5];
  const float* wqp   = (const float*)d_in[6];
  const float* wkp   = (const float*)d_in[7];
  const float* wvp   = (const float*)d_in[8];
  const float* gamma = (const float*)d_in[9];
  const float* wout  = (const float*)d_in[10];
  const float* wupd  = (const float*)d_in[11];
  const float* bupd  = (const float*)d_in[12];

  float* ws = (float*)d_ws;
  size_t cur = 0;
  auto alloc = [&](size_t n) { float* p = ws + cur; cur += n; return p; };

  float* s_all = alloc((size_t)BT_ * CS_);
  float* t_all = alloc((size_t)BT_ * 3);
  float* r_all = alloc((size_t)BT_ * 9);
  float* qb    = alloc((size_t)BT_ * H_ * C_);
  float* kb    = alloc((size_t)BT_ * H_ * C_);
  float* vb    = alloc((size_t)BT_ * H_ * C_);
  float* qp    = alloc((size_t)BT_ * H_ * PQ_ * 3);   // becomes qg in-place
  float* kp    = alloc((size_t)BT_ * H_ * PQ_ * 3);   // becomes kg in-place
  float* vp    = alloc((size_t)BT_ * H_ * PV_ * 3);   // becomes vg in-place
  float* sqb   = alloc((size_t)BT_ * H_);
  float* skb   = alloc((size_t)BT_ * H_);
  float* ob    = alloc((size_t)B_ * TP_ * H_ * C_);
  float* opgb  = alloc((size_t)B_ * TP_ * H_ * PV_ * 3);
  float* ocb   = alloc((size_t)B_ * TP_ * OCD_);
  float* sub   = alloc((size_t)B_ * TP_ * CS_);
  (void)ws_size; (void)in_sizes; (void)n_in; (void)out_size;

  prep_kernel<<<BT_, 128, 0, stream>>>(trans, rots, s, s_all, t_all, r_all);

  launch_gemm<192>(s_all, wq, qb, BT_, CS_, stream);
  launch_gemm<192>(s_all, wk, kb, BT_, CS_, stream);
  launch_gemm<192>(s_all, wv, vb, BT_, CS_, stream);
  launch_gemm<144>(s_all, wqp, qp, BT_, CS_, stream);
  launch_gemm<144>(s_all, wkp, kp, BT_, CS_, stream);
  launch_gemm<288>(s_all, wvp, vp, BT_, CS_, stream);

  int nbh = BT_ * H_;
  int thr = 256;
  to_global_kernel<<<(nbh + thr - 1) / thr, thr, 0, stream>>>(qp, PQ_, t_all, r_all, sqb);
  to_global_kernel<<<(nbh + thr - 1) / thr, thr, 0, stream>>>(kp, PQ_, t_all, r_all, skb);
  to_global_kernel<<<(nbh + thr - 1) / thr, thr, 0, stream>>>(vp, PV_, t_all, r_all, nullptr);

  attn_kernel<<<B_ * H_ * (TP_ / 16), 32, 0, stream>>>(qb, kb, vb, qp, kp, vp,
                                                       sqb, skb, gamma, ob, opgb);

  build_oc_kernel<<<(B_ * TP_ * H_ + 255) / 256, 256, 0, stream>>>(ob, opgb, t_all,
                                                                   r_all, ocb);
  launch_gemm<384>(ocb, wout, sub, B_ * TP_, OCD_, stream);

  final_kernel<<<B_ * TP_, 32, 0, stream>>>(sub, wupd, bupd, t_all, r_all,
                                            (float*)d_out);
}